// PhenoCapsuleHead_58033598104066
// MI455X (gfx1250) — compile-verified
//
#include <hip/hip_runtime.h>
#include <hip/hip_bf16.h>

typedef __attribute__((ext_vector_type(2))) float v2f;
typedef __attribute__((ext_vector_type(8))) float v8f;

#define D_IN   4096
#define N_OUT  2048   // N_PRIM * D_PRIM
#define N_PRIM 64
#define D_PRIM 32
#define N_CLS  25
#define D_CLS  16

// ---------------- Kernel 1: primary_pose = z @ W_prim + b_prim  (fp32 WMMA) ----
// Block tile 128x256, 8 waves, each wave computes 64x64 (4x4 WMMA frags).
// Double-buffered LDS (2 x 53KB) pipelines HBM fetch against WMMA compute.
#define BM 128
#define BN 256
#define BK 32
#define LDA (BM + 16)            // 144: +2-row K step shifts banks by 32 -> conflict-free
#define LDB (BN + 16)            // 272: same property
#define ASZ (BK * LDA)           // 4608 floats
#define BSZ (BK * LDB)           // 8704 floats
#define BUF (ASZ + BSZ)          // 13312 floats per buffer

__device__ __forceinline__
void load_tile(const float* __restrict__ A, const float* __restrict__ Bw,
               int m0, int n0, int k0, float* __restrict__ Asm,
               float* __restrict__ Bsm, int tid)
{
    // A tile 128x32, stored transposed Asm[k][m]
    #pragma unroll
    for (int i = 0; i < 4; ++i) {
        int idx = tid + i * 256;          // 0..1023
        int k4  = (idx & 7) * 4;          // 0..28
        int m   = idx >> 3;               // 0..127
        float4 v = *reinterpret_cast<const float4*>(
            &A[(size_t)(m0 + m) * D_IN + k0 + k4]);
        Asm[(k4 + 0) * LDA + m] = v.x;
        Asm[(k4 + 1) * LDA + m] = v.y;
        Asm[(k4 + 2) * LDA + m] = v.z;
        Asm[(k4 + 3) * LDA + m] = v.w;
    }
    // B tile 32x256, row-major Bsm[k][n]
    #pragma unroll
    for (int i = 0; i < 8; ++i) {
        int idx = tid + i * 256;          // 0..2047
        int n4  = (idx & 63) * 4;         // 0..252
        int kk  = idx >> 6;               // 0..31
        float4 v = *reinterpret_cast<const float4*>(
            &Bw[(size_t)(k0 + kk) * N_OUT + n0 + n4]);
        *reinterpret_cast<float4*>(&Bsm[kk * LDB + n4]) = v;
    }
}

__global__ __launch_bounds__(256)
void k_primary(const float* __restrict__ A,   // z      [4096,4096]
               const float* __restrict__ Bw,  // W_prim [4096,2048]
               const float* __restrict__ bias,// b_prim [2048]
               float* __restrict__ C)         // out    [4096,2048]
{
    extern __shared__ float smem[];           // 2 * BUF floats (104 KB)

    const int tid  = threadIdx.x;
    const int m0   = blockIdx.y * BM;
    const int n0   = blockIdx.x * BN;
    const int wave = tid >> 5;
    const int lane = tid & 31;
    const int half = lane >> 4;               // 0: K pair {0,1}, 1: K pair {2,3}
    const int lm   = lane & 15;               // M (A) / N (B) index within fragment
    const int wm   = (wave & 1) * 64;         // wave row offset in block tile
    const int wn   = (wave >> 1) * 64;        // wave col offset in block tile

    v8f acc[4][4];
    #pragma unroll
    for (int i = 0; i < 4; ++i)
        #pragma unroll
        for (int j = 0; j < 4; ++j)
            acc[i][j] = (v8f){0.f,0.f,0.f,0.f,0.f,0.f,0.f,0.f};

    const int nblk = D_IN / BK;               // 128
    load_tile(A, Bw, m0, n0, 0, smem, smem + ASZ, tid);
    __syncthreads();

    for (int kb = 0; kb < nblk; ++kb) {
        const float* As = smem + (kb & 1) * BUF;
        const float* Bs = As + ASZ;
        if (kb + 1 < nblk) {
            float* An = smem + ((kb + 1) & 1) * BUF;
            load_tile(A, Bw, m0, n0, (kb + 1) * BK, An, An + ASZ, tid);
        }
        #pragma unroll
        for (int kk = 0; kk < BK; kk += 4) {
            const int krow = kk + 2 * half;
            v2f a[4], b[4];
            #pragma unroll
            for (int i = 0; i < 4; ++i) {
                a[i].x = As[(krow + 0) * LDA + wm + i * 16 + lm];
                a[i].y = As[(krow + 1) * LDA + wm + i * 16 + lm];
            }
            #pragma unroll
            for (int j = 0; j < 4; ++j) {
                b[j].x = Bs[(krow + 0) * LDB + wn + j * 16 + lm];
                b[j].y = Bs[(krow + 1) * LDB + wn + j * 16 + lm];
            }
            #pragma unroll
            for (int i = 0; i < 4; ++i)
                #pragma unroll
                for (int j = 0; j < 4; ++j)
                    acc[i][j] = __builtin_amdgcn_wmma_f32_16x16x4_f32(
                        false, a[i], false, b[j], (short)0, acc[i][j], false, false);
        }
        __syncthreads();
    }

    // epilogue: C layout per ISA (VGPR r: lanes 0-15 -> M=r, lanes 16-31 -> M=r+8)
    #pragma unroll
    for (int i = 0; i < 4; ++i) {
        #pragma unroll
        for (int j = 0; j < 4; ++j) {
            const int col   = n0 + wn + j * 16 + lm;
            const float bv  = bias[col];
            const int rbase = m0 + wm + i * 16 + half * 8;
            #pragma unroll
            for (int r = 0; r < 8; ++r)
                C[(size_t)(rbase + r) * N_OUT + col] = acc[i][j][r] + bv;
        }
    }
}

// ---------------- Kernel 2: primary_act = clamp(sigmoid(LN(z) @ W_act + b_act)) --
__global__ __launch_bounds__(256)
void k_act(const float* __restrict__ z, const float* __restrict__ g,
           const float* __restrict__ bb, const float* __restrict__ W,
           const float* __restrict__ ba, float* __restrict__ out_act)
{
    __shared__ float zs[D_IN];
    __shared__ float red[256];
    __shared__ float part[4][64];

    const int b = blockIdx.x, t = threadIdx.x;
    const float* zr = z + (size_t)b * D_IN;

    #pragma unroll
    for (int i = 0; i < 4; ++i) {
        float4 v = *reinterpret_cast<const float4*>(&zr[(t + i * 256) * 4]);
        *reinterpret_cast<float4*>(&zs[(t + i * 256) * 4]) = v;
    }
    __syncthreads();

    float s = 0.f;
    for (int i = t; i < D_IN; i += 256) s += zs[i];
    red[t] = s; __syncthreads();
    for (int off = 128; off; off >>= 1) { if (t < off) red[t] += red[t + off]; __syncthreads(); }
    const float mu = red[0] * (1.f / D_IN);
    __syncthreads();

    float s2 = 0.f;
    for (int i = t; i < D_IN; i += 256) { float d = zs[i] - mu; s2 += d * d; }
    red[t] = s2; __syncthreads();
    for (int off = 128; off; off >>= 1) { if (t < off) red[t] += red[t + off]; __syncthreads(); }
    const float rinv = rsqrtf(red[0] * (1.f / D_IN) + 1e-5f);
    __syncthreads();

    for (int i = t; i < D_IN; i += 256) zs[i] = (zs[i] - mu) * rinv * g[i] + bb[i];
    __syncthreads();

    const int j = t & 63, p = t >> 6;
    float acc = 0.f;
    for (int i = p * 1024; i < (p + 1) * 1024; ++i)
        acc += zs[i] * W[(size_t)i * N_PRIM + j];
    part[p][j] = acc; __syncthreads();

    if (t < 64) {
        float v = part[0][t] + part[1][t] + part[2][t] + part[3][t] + ba[t];
        float sg = 1.f / (1.f + expf(-v));
        out_act[(size_t)b * N_PRIM + t] = fminf(fmaxf(sg, 0.f), 1.f);
    }
}

// ---------------- Kernel 3: votes + EM routing + logit head, all in LDS --------
__global__ __launch_bounds__(512)
void k_routing(const float* __restrict__ ppose,  // [B,64,32]
               const float* __restrict__ pact,   // [B,64]
               const float* __restrict__ Wc,     // [64,32,25,16]
               const float* __restrict__ g2, const float* __restrict__ b2,
               const float* __restrict__ Wo, const float* __restrict__ bo,
               float* __restrict__ out_logits, float* __restrict__ out_act,
               float* __restrict__ out_pose, float* __restrict__ out_q)
{
    extern __shared__ float sm[];
    float* pp    = sm;              // 2048
    float* votes = pp + 2048;       // 25600  (100 KB: fits in 320 KB WGP LDS)
    float* a     = votes + 25600;   // 64
    float* qa    = a + 64;          // 1600
    float* ql    = qa + 1600;       // 1600 (logits, then q)
    float* pose  = ql + 1600;       // 400
    float* act   = pose + 400;      // 25
    float* red   = act + 25;        // 1

    const int b = blockIdx.x, t = threadIdx.x;

    { float4 v = *reinterpret_cast<const float4*>(&ppose[(size_t)b * 2048 + t * 4]);
      *reinterpret_cast<float4*>(&pp[t * 4]) = v; }
    if (t < 64) a[t] = pact[(size_t)b * 64 + t];
    __syncthreads();

    // votes[n, m*16+o] = sum_d pp[n,d] * Wc[n,d,m,o]
    for (int idx = t; idx < 25600; idx += 512) {
        int n = idx / 400, off = idx % 400;
        const float* w = Wc + (size_t)n * 12800 + off;   // stride 400 over d
        const float* p = pp + n * 32;
        float acc = 0.f;
        #pragma unroll
        for (int d = 0; d < 32; ++d) acc += p[d] * w[(size_t)d * 400];
        votes[idx] = acc;
    }
    __syncthreads();

    // iteration 0: uniform routing
    for (int idx = t; idx < 1600; idx += 512) qa[idx] = a[idx / 25] * (1.f / N_CLS);
    if (t == 0) { float s = 0.f; for (int n = 0; n < 64; ++n) s += a[n]; red[0] = s * (1.f / N_CLS); }
    __syncthreads();
    if (t < N_CLS) act[t] = 1.f / (1.f + expf(-red[0]));
    for (int idx = t; idx < 400; idx += 512) {
        const int m = idx >> 4;
        float acc = 0.f;
        for (int n = 0; n < 64; ++n) acc += qa[n * 25 + m] * votes[n * 400 + idx];
        pose[idx] = acc;
    }
    __syncthreads();

    const float scale = 0.25f;  // 1/sqrt(D_CLS)
    for (int it = 0; it < 2; ++it) {
        // agreement logits
        for (int idx = t; idx < 1600; idx += 512) {
            const int n = idx / 25, m = idx % 25;
            const float* v  = votes + n * 400 + m * 16;
            const float* po = pose + m * 16;
            float acc = 0.f;
            #pragma unroll
            for (int o = 0; o < 16; ++o) acc += v[o] * po[o];
            ql[idx] = acc * scale + logf(act[m] + 1e-8f);
        }
        __syncthreads();
        // softmax over M per input capsule; q and qa
        if (t < 64) {
            float mx = -1e30f;
            #pragma unroll
            for (int m = 0; m < 25; ++m) mx = fmaxf(mx, ql[t * 25 + m]);
            float e[25], s = 0.f;
            #pragma unroll
            for (int m = 0; m < 25; ++m) { e[m] = expf(ql[t * 25 + m] - mx); s += e[m]; }
            const float inv = 1.f / s, av = a[t];
            #pragma unroll
            for (int m = 0; m < 25; ++m) { float qv = e[m] * inv; ql[t * 25 + m] = qv; qa[t * 25 + m] = qv * av; }
        }
        __syncthreads();
        // pose & act update
        for (int idx = t; idx < 400; idx += 512) {
            const int m = idx >> 4;
            float acc = 0.f;
            for (int n = 0; n < 64; ++n) acc += qa[n * 25 + m] * votes[n * 400 + idx];
            pose[idx] = acc;
        }
        if (t < N_CLS) {
            float s = 0.f;
            for (int n = 0; n < 64; ++n) s += qa[n * 25 + t];
            act[t] = 1.f / (1.f + expf(-s));
        }
        __syncthreads();
    }

    // outputs
    for (int idx = t; idx < 1600; idx += 512) out_q[(size_t)b * 1600 + idx] = ql[idx];
    for (int idx = t; idx < 400;  idx += 512) out_pose[(size_t)b * 400 + idx] = pose[idx];
    if (t < N_CLS) {
        out_act[(size_t)b * 25 + t] = act[t];
        const float* po = pose + t * 16;
        float mu = 0.f;
        #pragma unroll
        for (int o = 0; o < 16; ++o) mu += po[o];
        mu *= (1.f / 16.f);
        float var = 0.f;
        #pragma unroll
        for (int o = 0; o < 16; ++o) { float d = po[o] - mu; var += d * d; }
        var *= (1.f / 16.f);
        const float rinv = rsqrtf(var + 1e-5f);
        float acc = 0.f;
        #pragma unroll
        for (int o = 0; o < 16; ++o) acc += ((po[o] - mu) * rinv * g2[o] + b2[o]) * Wo[o];
        out_logits[(size_t)b * 25 + t] = acc + bo[0];
    }
}

// ---------------- host launch ---------------------------------------------------
extern "C" void kernel_launch(void* const* d_in, const int* in_sizes, int n_in,
                              void* d_out, int out_size, void* d_ws, size_t ws_size,
                              hipStream_t stream) {
    const float* z      = (const float*)d_in[0];
    const float* W_prim = (const float*)d_in[1];
    const float* b_prim = (const float*)d_in[2];
    const float* ln1_g  = (const float*)d_in[3];
    const float* ln1_b  = (const float*)d_in[4];
    const float* W_act  = (const float*)d_in[5];
    const float* b_act  = (const float*)d_in[6];
    const float* W_caps = (const float*)d_in[7];
    const float* ln2_g  = (const float*)d_in[8];
    const float* ln2_b  = (const float*)d_in[9];
    const float* W_out  = (const float*)d_in[10];
    const float* b_out  = (const float*)d_in[11];

    float* out = (float*)d_out;
    // return order: class_logits, act, pose, q, primary_pose, primary_act
    float* o_logits = out;                 // 4096*25
    float* o_act    = out + 102400;        // 4096*25
    float* o_pose   = out + 204800;        // 4096*25*16
    float* o_q      = out + 1843200;       // 4096*64*25
    float* o_ppose  = out + 8396800;       // 4096*2048
    float* o_pact   = out + 16785408;      // 4096*64

    dim3 g1(N_OUT / BN, 4096 / BM);        // (8, 32)
    const size_t sh1 = (size_t)2 * BUF * sizeof(float);  // 104 KB double buffer
    k_primary<<<g1, 256, sh1, stream>>>(z, W_prim, b_prim, o_ppose);

    k_act<<<4096, 256, 0, stream>>>(z, ln1_g, ln1_b, W_act, b_act, o_pact);

    const size_t sh3 = (size_t)(2048 + 25600 + 64 + 1600 + 1600 + 400 + 25 + 1) * sizeof(float);
    k_routing<<<4096, 512, sh3, stream>>>(o_ppose, o_pact, W_caps, ln2_g, ln2_b,
                                          W_out, b_out,
                                          o_logits, o_act, o_pose, o_q);
}